// LTCCell_60730837565793
// MI455X (gfx1250) — compile-verified
//
#include <hip/hip_runtime.h>
#include <hip/hip_bf16.h>

#define B_   128
#define I_   128
#define U_   512
#define NUNF 6
#define TJ   8            // rows of packed params per TDM tile
#define NT   (U_ / TJ)    // 64 tiles per unfold
#define NBUF 3            // triple-buffered TDM pipeline

typedef unsigned int v4u __attribute__((ext_vector_type(4)));
typedef int          v4i __attribute__((ext_vector_type(4)));
typedef int          v8i __attribute__((ext_vector_type(8)));

// ---- gfx1250 fast tanh (new trans op on CDNA5); safe fallback ----
__device__ __forceinline__ float fast_tanh(float x) {
#if defined(__AMDGCN__) && __has_builtin(__builtin_amdgcn_tanhf)
    return __builtin_amdgcn_tanhf(x);
#else
    return tanhf(x);
#endif
}

__device__ __forceinline__ int uniform_tid() {
#if defined(__AMDGCN__)
    return __builtin_amdgcn_readfirstlane((int)threadIdx.x);
#else
    return 0;
#endif
}

__device__ __forceinline__ void wait_tensorcnt_le(int n) {
#if defined(__AMDGCN__)
    if (n) __builtin_amdgcn_s_wait_tensorcnt(1);
    else   __builtin_amdgcn_s_wait_tensorcnt(0);
#endif
}

// TDM: stream TJ consecutive rows (each row = 512 float4 = 1024 x 8B units) of the
// packed [U_ x U_] float4 parameter tensor into LDS. D# per CDNA5 ISA §8.
__device__ __forceinline__ void tdm_load_rows(const float4* gbase, unsigned lds_byte_addr, int j0) {
#if defined(__AMDGCN__)
    unsigned long long ga = (unsigned long long)(const void*)(gbase + (size_t)j0 * U_);
    const unsigned row_u64 = U_ * 2u;   // 1024 8-byte units per row

    v4u g0;
    g0[0] = 1u;                                   // count=1, user mode, gather off
    g0[1] = lds_byte_addr;                        // lds_addr
    g0[2] = (unsigned)ga;                         // global_addr[31:0]
    g0[3] = (unsigned)((ga >> 32) & 0x1FFFFFFull) // global_addr[56:32]
          | (2u << 30);                           // type=2 ("image")

    v8i g1;
    g1[0] = (int)(3u << 16);                      // workgroup_mask=0, data_size=8B
    g1[1] = (int)((row_u64 & 0xFFFFu) << 16);     // tensor_dim0[15:0]
    g1[2] = (int)(((row_u64 >> 16) & 0xFFFFu)     // tensor_dim0[31:16]
          | ((unsigned)U_ << 16));                // tensor_dim1[15:0] = 512
    g1[3] = (int)((row_u64 & 0xFFFFu) << 16);     // tensor_dim1[31:16]=0, tile_dim0=1024
    g1[4] = (int)(TJ);                            // tile_dim1=TJ, tile_dim2=0
    g1[5] = (int)row_u64;                         // tensor_dim0_stride[31:0]
    g1[6] = 0;                                    // stride0[47:32]=0, stride1[15:0]=0
    g1[7] = 0;

    v4i z  = {0, 0, 0, 0};
#if __clang_major__ >= 23
    v8i z8 = {0, 0, 0, 0, 0, 0, 0, 0};
    __builtin_amdgcn_tensor_load_to_lds(g0, g1, z, z, z8, 0);
#else
    __builtin_amdgcn_tensor_load_to_lds(g0, g1, z, z, 0);
#endif
#else
    (void)gbase; (void)lds_byte_addr; (void)j0;
#endif
}

// ---------------- Kernel: 6 semi-implicit unfolds (one WG per batch row) ----------------
// First in the module so the disasm snippet shows the hot loop.
__global__ __launch_bounds__(U_) void ltc_unfold(const float4* __restrict__ pk,
                                                 const float* __restrict__ num_s_g,
                                                 const float* __restrict__ den_s_g,
                                                 const float* __restrict__ state,
                                                 const float* __restrict__ vleak,
                                                 const float* __restrict__ gleak,
                                                 const float* __restrict__ cm_t,
                                                 float* __restrict__ out) {
    __shared__ __align__(16) float4 tiles[NBUF][TJ * U_];  // 3 x 64 KB TDM staging buffers
    __shared__ float vsh[U_];                              // state broadcast buffer

    const int b = blockIdx.x;
    const int u = threadIdx.x;

    float v = state[u];
    vsh[u] = v;
    const float num0 = num_s_g[b * U_ + u];
    const float den0 = den_s_g[b * U_ + u];
    const float cm   = cm_t[u];
    const float gl   = gleak[u];
    const float glvl = gl * vleak[u];

    unsigned ldsb[NBUF];
    #pragma unroll
    for (int q = 0; q < NBUF; ++q)
        ldsb[q] = (unsigned)(unsigned long long)(const void*)&tiles[q][0];

    const bool issuer = (uniform_tid() == 0);   // scalar branch: only wave 0 drives the TDM
    __syncthreads();

    for (int t = 0; t < NUNF; ++t) {
        float an = num0, ad = den0;
        if (issuer) {                                     // prime: two tiles in flight
            tdm_load_rows(pk, ldsb[0], 0);
            tdm_load_rows(pk, ldsb[1], TJ);
        }
        for (int jt = 0; jt < NT; ++jt) {
            if (issuer) {
                if (jt + 1 < NT) wait_tensorcnt_le(1);    // tile jt landed (in-order TDM)
                else             wait_tensorcnt_le(0);
            }
            __syncthreads();   // publishes tile jt ready AND all waves done with tile jt-1
            if (issuer && (jt + 2 < NT))
                tdm_load_rows(pk, ldsb[(jt + 2) % NBUF], (jt + 2) * TJ);

            const float4* bp = tiles[jt % NBUF];
            const int j0 = jt * TJ;
            #pragma unroll
            for (int jj = 0; jj < TJ; ++jj) {
                float  vj = vsh[j0 + jj];                 // LDS broadcast (conflict-free)
                float4 p  = bp[jj * U_ + u];              // ds_load_b128, bank-friendly
                float  a  = p.y * (vj - p.x);             // 0.5*sigma*(v - mu)
                float  s  = fmaf(0.5f, fast_tanh(a), 0.5f);
                ad = fmaf(p.z, s, ad);                    // + W*sig
                an = fmaf(p.w, s, an);                    // + W*erev*sig
            }
        }
        __syncthreads();                                  // all reads of vsh/tiles complete
        v = (fmaf(cm, v, glvl) + an) / (cm + gl + ad);
        vsh[u] = v;
        __syncthreads();                                  // new state visible to all waves
    }
    out[b * U_ + u] = v;
}

// ---------------- Kernel: pack (mu, 0.5*sigma, W, W*erev) as float4 ----------------
__global__ __launch_bounds__(256) void ltc_pack(const float* __restrict__ mu,
                                                const float* __restrict__ sg,
                                                const float* __restrict__ W,
                                                const float* __restrict__ er,
                                                float4* __restrict__ pk) {
    int idx = blockIdx.x * blockDim.x + threadIdx.x;
    if (idx < U_ * U_) {
        float w = W[idx];
        pk[idx] = make_float4(mu[idx], 0.5f * sg[idx], w, w * er[idx]);
    }
}

// ---------------- Kernel: sensory sums (one WG per batch row) ----------------
__global__ __launch_bounds__(U_) void ltc_sensory(const float* __restrict__ inputs,
                                                  const float* __restrict__ iw,
                                                  const float* __restrict__ ib,
                                                  const float* __restrict__ smu,
                                                  const float* __restrict__ ssg,
                                                  const float* __restrict__ sW,
                                                  const float* __restrict__ ser,
                                                  float* __restrict__ num_out,
                                                  float* __restrict__ den_out) {
    __shared__ float xs[I_];
    const int b = blockIdx.x;
    const int u = threadIdx.x;
    if (u < I_) {
        int t = b * I_ + u;
        xs[u] = fmaf(inputs[t], iw[t], ib[t]);
    }
    __syncthreads();
    float num = 0.f, den = 0.f;
    #pragma unroll 4
    for (int i = 0; i < I_; ++i) {
        int t = i * U_ + u;                           // coalesced across the wave
        float a   = 0.5f * ssg[t] * (xs[i] - smu[t]); // xs[i]: LDS broadcast
        float s   = fmaf(0.5f, fast_tanh(a), 0.5f);   // sigmoid via v_tanh_f32
        float act = sW[t] * s;
        num = fmaf(act, ser[t], num);
        den += act;
    }
    num_out[b * U_ + u] = num;
    den_out[b * U_ + u] = den;
}

extern "C" void kernel_launch(void* const* d_in, const int* in_sizes, int n_in,
                              void* d_out, int out_size, void* d_ws, size_t ws_size,
                              hipStream_t stream) {
    const float* inputs = (const float*)d_in[0];
    const float* state  = (const float*)d_in[1];
    const float* iw     = (const float*)d_in[2];
    const float* ib     = (const float*)d_in[3];
    const float* smu    = (const float*)d_in[4];
    const float* ssg    = (const float*)d_in[5];
    const float* sW     = (const float*)d_in[6];
    const float* ser    = (const float*)d_in[7];
    const float* mu     = (const float*)d_in[8];
    const float* sg     = (const float*)d_in[9];
    const float* W      = (const float*)d_in[10];
    const float* er     = (const float*)d_in[11];
    const float* vleak  = (const float*)d_in[12];
    const float* gleak  = (const float*)d_in[13];
    const float* cmt    = (const float*)d_in[14];
    float* out = (float*)d_out;

    // workspace layout: [num_sens 256KB][den_sens 256KB][packed params 4MB]
    float*  num_s = (float*)d_ws;
    float*  den_s = num_s + B_ * U_;
    float4* pk    = (float4*)((char*)d_ws + 2ull * B_ * U_ * sizeof(float));

    ltc_pack   <<<(U_ * U_ + 255) / 256, 256, 0, stream>>>(mu, sg, W, er, pk);
    ltc_sensory<<<B_, U_, 0, stream>>>(inputs, iw, ib, smu, ssg, sW, ser, num_s, den_s);
    ltc_unfold <<<B_, U_, 0, stream>>>(pk, num_s, den_s, state, vleak, gleak, cmt, out);
}